// GATModel_4741643894830
// MI455X (gfx1250) — compile-verified
//
#include <hip/hip_runtime.h>
#include <cstdint>
#include <cstddef>

// ---------------------------------------------------------------------------
// 2-layer GAT for MI455X (gfx1250, wave32).
// GEMMs use V_WMMA_F32_16X16X4_F32 (full-precision fp32 WMMA).
//   - fragments loaded branch-free (zero-padded / pair-interleaved operands)
//   - gemm2 stages W2 tiles in LDS (ds_load_b64 B-fragments, 8-way reuse)
//   - uniform full-tile store guards hoisted out of the loops
// Edge softmax/aggregation uses float global atomics (max via int trick, add);
// edge kernels are templated on <H, C> so divisions become shifts and the
// per-lane channel loops fully unroll.
// ---------------------------------------------------------------------------

typedef __attribute__((ext_vector_type(2))) float v2f;
typedef __attribute__((ext_vector_type(8))) float v8f;

#define NEG_SLOPE 0.2f

__device__ __forceinline__ void atomicMaxF(float* addr, float v) {
    if (v >= 0.0f) atomicMax((int*)addr, __float_as_int(v));
    else           atomicMin((unsigned int*)addr, (unsigned int)__float_as_int(v));
}

__device__ __forceinline__ void edge_sd(const int* __restrict__ ei, int E, int e,
                                        int& s, int& d) {
    if (e < E) { s = ei[e]; d = ei[E + e]; }   // ei: [2, E] row-major
    else       { s = e - E; d = s; }           // appended self loops
}

__device__ __forceinline__ constexpr int ilog2(int v) {
    int r = 0; while (v > 1) { v >>= 1; ++r; } return r;
}

// ------------------------------- fills -------------------------------------
__global__ void fill_k(float* __restrict__ p, size_t n, float v) {
    size_t i = (size_t)blockIdx.x * blockDim.x + threadIdx.x;
    size_t stride = (size_t)gridDim.x * blockDim.x;
    for (; i < n; i += stride) p[i] = v;
}

// ----------------- operand packing (zero-pad K=3 -> K=4) -------------------
__global__ void pack_x4_k(const float* __restrict__ x, float* __restrict__ x4, int N) {
    int i = blockIdx.x * blockDim.x + threadIdx.x;
    int total = N * 4, stride = gridDim.x * blockDim.x;
    for (; i < total; i += stride) {
        int n = i >> 2, k = i & 3;
        x4[i] = (k < 3) ? x[n * 3 + k] : 0.f;
    }
}
// W1p pair-interleaved: W1p[(k/2)*512 + n*2 + (k&1)] = k<3 ? W1[k*256+n] : 0
__global__ void pack_w1_k(const float* __restrict__ W1, float* __restrict__ W1p) {
    int i = blockIdx.x * blockDim.x + threadIdx.x;
    if (i < 1024) {
        int k = i >> 8, n = i & 255;
        W1p[(k >> 1) * 512 + n * 2 + (k & 1)] = (k < 3) ? W1[k * 256 + n] : 0.f;
    }
}

// ---------------------- Layer-1 GEMM: x[N,3] @ W1[3,256] -------------------
__global__ void gemm1_wmma(const float* __restrict__ x4, const float* __restrict__ W1p,
                           float* __restrict__ h1, int N) {
    const int lane = threadIdx.x & 31;
    const int wave = threadIdx.x >> 5;          // 0..7
    const int mt   = blockIdx.x;
    const int mrow = mt * 16 + (lane & 15);
    const int mc   = mrow < N ? mrow : N - 1;   // clamp; OOB rows masked at store
    const int koff = (lane >> 4) << 1;          // 0 or 2
    const int mbase = mt * 16 + ((lane >> 4) << 3);

    const v2f a = *(const v2f*)&x4[(size_t)mc * 4 + koff];

    if ((mt * 16 + 16) <= N) {                  // uniform: full tile (common)
        for (int t = wave; t < 16; t += 8) {    // 16 N-tiles of 256 cols
            const int n = t * 16 + (lane & 15);
            const v2f b = *(const v2f*)&W1p[(koff >> 1) * 512 + n * 2];
            v8f c = {};
            c = __builtin_amdgcn_wmma_f32_16x16x4_f32(false, a, false, b,
                                                      (short)0, c, false, false);
            #pragma unroll
            for (int r = 0; r < 8; ++r)
                h1[(size_t)(mbase + r) * 256 + n] = c[r];
        }
    } else {                                    // uniform: ragged tail tile
        for (int t = wave; t < 16; t += 8) {
            const int n = t * 16 + (lane & 15);
            const v2f b = *(const v2f*)&W1p[(koff >> 1) * 512 + n * 2];
            v8f c = {};
            c = __builtin_amdgcn_wmma_f32_16x16x4_f32(false, a, false, b,
                                                      (short)0, c, false, false);
            #pragma unroll
            for (int r = 0; r < 8; ++r)
                if (mbase + r < N) h1[(size_t)(mbase + r) * 256 + n] = c[r];
        }
    }
}

// ---------------- Layer-2 GEMM: A[N,256] @ W2[256,128] ---------------------
#define TILE_K 64
__global__ void gemm2_wmma(const float* __restrict__ A, const float* __restrict__ B,
                           float* __restrict__ D, int N) {
    __shared__ float Bs[(TILE_K / 2) * 256];    // 32 KB: [k/2][n*2 + (k&1)]
    const int tid  = threadIdx.x;
    const int lane = tid & 31;
    const int wave = tid >> 5;                  // n-tile 0..7
    const int mt   = blockIdx.x;
    const int mrow = mt * 16 + (lane & 15);
    const int mc   = mrow < N ? mrow : N - 1;
    const int koff = (lane >> 4) << 1;          // 0 or 2
    const int n    = wave * 16 + (lane & 15);
    const float* arow = A + (size_t)mc * 256;

    v8f c = {};
    for (int kb = 0; kb < 256; kb += TILE_K) {
        __syncthreads();                        // prev chunk fully consumed
        // cooperative load: B[kb..kb+63][0..127] -> Bs (pair-interleaved)
        for (int i = tid * 4; i < TILE_K * 128; i += 256 * 4) {
            const int k  = i >> 7;              // local row 0..63
            const int n0 = i & 127;
            const float4 w = *(const float4*)&B[(size_t)(kb + k) * 128 + n0];
            const int base = (k >> 1) * 256 + (k & 1);
            Bs[base + (n0 + 0) * 2] = w.x;
            Bs[base + (n0 + 1) * 2] = w.y;
            Bs[base + (n0 + 2) * 2] = w.z;
            Bs[base + (n0 + 3) * 2] = w.w;
        }
        __syncthreads();
        #pragma unroll
        for (int kk = 0; kk < TILE_K; kk += 4) {
            const v2f a = *(const v2f*)(arow + kb + kk + koff);
            const v2f b = *(const v2f*)&Bs[((kk + koff) >> 1) * 256 + n * 2];
            c = __builtin_amdgcn_wmma_f32_16x16x4_f32(false, a, false, b,
                                                      (short)0, c, false, false);
        }
    }
    const int mbase = mt * 16 + ((lane >> 4) << 3);
    if ((mt * 16 + 16) <= N) {                  // uniform guard, outside K-loop
        #pragma unroll
        for (int r = 0; r < 8; ++r)
            D[(size_t)(mbase + r) * 128 + n] = c[r];
    } else {
        #pragma unroll
        for (int r = 0; r < 8; ++r)
            if (mbase + r < N) D[(size_t)(mbase + r) * 128 + n] = c[r];
    }
}

// -------- per-(node,head) attention dots: a_src = <h, att_src>, etc. -------
template<int H, int C>
__global__ void att_dots_k(const float* __restrict__ feat,
                           const float* __restrict__ attS, const float* __restrict__ attD,
                           float* __restrict__ as, float* __restrict__ ad, int N) {
    constexpr int LH = ilog2(H);
    int tid = blockIdx.x * blockDim.x + threadIdx.x;
    int total = N * H;
    int stride = gridDim.x * blockDim.x;
    for (int i = tid; i < total; i += stride) {
        int n = i >> LH, h = i & (H - 1);
        const float* f = feat + (size_t)n * H * C + (size_t)h * C;
        float s = 0.f, d = 0.f;
        #pragma unroll 4
        for (int cidx = 0; cidx < C; ++cidx) {
            float fv = f[cidx];
            s += fv * attS[h * C + cidx];
            d += fv * attD[h * C + cidx];
        }
        as[i] = s; ad[i] = d;
    }
}

// ---------------- edge pass 1: segment max of leaky-relu(e) ----------------
template<int H>
__global__ void edge_max_k(const int* __restrict__ ei, int E, int Etot,
                           const float* __restrict__ as, const float* __restrict__ ad,
                           float* __restrict__ emax) {
    constexpr int LH = ilog2(H);
    int tid = blockIdx.x * blockDim.x + threadIdx.x;
    int total = Etot * H;
    int stride = gridDim.x * blockDim.x;
    for (int i = tid; i < total; i += stride) {
        int e = i >> LH, h = i & (H - 1);
        int s, d; edge_sd(ei, E, e, s, d);
        float v = as[s * H + h] + ad[d * H + h];
        v = v > 0.f ? v : NEG_SLOPE * v;
        atomicMaxF(&emax[d * H + h], v);
    }
}

// ---------------- edge pass 2: denom = segment sum of exp(e-max) -----------
template<int H>
__global__ void edge_den_k(const int* __restrict__ ei, int E, int Etot,
                           const float* __restrict__ as, const float* __restrict__ ad,
                           const float* __restrict__ emax, float* __restrict__ den) {
    constexpr int LH = ilog2(H);
    int tid = blockIdx.x * blockDim.x + threadIdx.x;
    int total = Etot * H;
    int stride = gridDim.x * blockDim.x;
    for (int i = tid; i < total; i += stride) {
        int e = i >> LH, h = i & (H - 1);
        int s, d; edge_sd(ei, E, e, s, d);
        float v = as[s * H + h] + ad[d * H + h];
        v = v > 0.f ? v : NEG_SLOPE * v;
        atomicAdd(&den[d * H + h], __expf(v - emax[d * H + h]));
    }
}

// ---------- edge pass 3: agg[dst] += alpha * feat[src] (wave/edge) ---------
template<int H, int C>
__global__ void edge_agg_k(const int* __restrict__ ei, int E, int Etot,
                           const float* __restrict__ as, const float* __restrict__ ad,
                           const float* __restrict__ emax, const float* __restrict__ den,
                           const float* __restrict__ feat, float* __restrict__ agg) {
    constexpr int LC = ilog2(C);
    constexpr int HC = H * C;
    int gt = blockIdx.x * blockDim.x + threadIdx.x;
    int lane   = gt & 31;
    int wave   = gt >> 5;
    int nwaves = (gridDim.x * blockDim.x) >> 5;
    for (int e = wave; e < Etot; e += nwaves) {
        int s, d; edge_sd(ei, E, e, s, d);
        #pragma unroll
        for (int j = 0; j < HC / 32; ++j) {
            const int fc = lane + j * 32;
            const int h = fc >> LC;
            float v = as[s * H + h] + ad[d * H + h];
            v = v > 0.f ? v : NEG_SLOPE * v;
            float alpha = __expf(v - emax[d * H + h]) / (den[d * H + h] + 1e-16f);
            atomicAdd(&agg[(size_t)d * HC + fc], alpha * feat[(size_t)s * HC + fc]);
        }
    }
}

// -------------------- bias + ELU (layer-1 activation) ----------------------
__global__ void elu_bias_k(float* __restrict__ a, const float* __restrict__ b,
                           int N, int F) {
    size_t i = (size_t)blockIdx.x * blockDim.x + threadIdx.x;
    size_t total = (size_t)N * F;
    size_t stride = (size_t)gridDim.x * blockDim.x;
    for (; i < total; i += stride) {
        float v = a[i] + b[i & (size_t)(F - 1)];   // F is a power of two
        a[i] = v > 0.f ? v : (__expf(v) - 1.f);
    }
}

// ------------------------------ final bias ---------------------------------
__global__ void bias_add_k(float* __restrict__ out, const float* __restrict__ b,
                           int N, int F) {
    size_t i = (size_t)blockIdx.x * blockDim.x + threadIdx.x;
    size_t total = (size_t)N * F;
    size_t stride = (size_t)gridDim.x * blockDim.x;
    for (; i < total; i += stride) out[i] += b[i & (size_t)(F - 1)];
}

// ---------------------------------------------------------------------------
extern "C" void kernel_launch(void* const* d_in, const int* in_sizes, int n_in,
                              void* d_out, int out_size, void* d_ws, size_t ws_size,
                              hipStream_t stream) {
    const float* x    = (const float*)d_in[0];   // [N,3]
    const int*   ei   = (const int*)  d_in[1];   // [2,E] int32 (JAX default x64-off)
    const float* W1   = (const float*)d_in[2];   // [3,256]
    const float* aS1  = (const float*)d_in[3];   // [4,64]
    const float* aD1  = (const float*)d_in[4];   // [4,64]
    const float* b1   = (const float*)d_in[5];   // [256]
    const float* W2   = (const float*)d_in[6];   // [256,128]
    const float* aS2  = (const float*)d_in[7];   // [1,128]
    const float* aD2  = (const float*)d_in[8];   // [1,128]
    const float* b2   = (const float*)d_in[9];   // [128]
    float* out = (float*)d_out;                  // [N,128]

    const int N    = in_sizes[0] / 3;
    const int E    = in_sizes[1] / 2;
    const int Etot = E + N;                      // with self loops

    // ---- workspace layout (floats) ----
    float* ws    = (float*)d_ws;
    float* h1    = ws;                                // N*256
    float* a1s   = h1    + (size_t)N * 256;           // N*4
    float* a1d   = a1s   + (size_t)N * 4;             // N*4
    float* emax1 = a1d   + (size_t)N * 4;             // N*4
    float* den1  = emax1 + (size_t)N * 4;             // N*4
    float* agg1  = den1  + (size_t)N * 4;             // N*256
    float* h2    = agg1  + (size_t)N * 256;           // N*128
    float* a2s   = h2    + (size_t)N * 128;           // N
    float* a2d   = a2s   + (size_t)N;                 // N
    float* emax2 = a2d   + (size_t)N;                 // N
    float* den2  = emax2 + (size_t)N;                 // N
    float* x4    = den2  + (size_t)N;                 // N*4 (padded x)
    float* W1p   = x4    + (size_t)N * 4;             // 1024 (padded W1)

    const int TB = 256;
    const int mtiles = (N + 15) / 16;
    const float NEG_INF = -__builtin_huge_valf();

    // ---- init accumulators (den1..agg1 contiguous -> one zero fill) ----
    fill_k<<<512, TB, 0, stream>>>(emax1, (size_t)N * 4, NEG_INF);
    fill_k<<<2048, TB, 0, stream>>>(den1, (size_t)N * (4 + 256), 0.f);
    fill_k<<<512, TB, 0, stream>>>(emax2, (size_t)N, NEG_INF);
    fill_k<<<512, TB, 0, stream>>>(den2, (size_t)N, 0.f);
    fill_k<<<2048, TB, 0, stream>>>(out, (size_t)N * 128, 0.f);

    // ---- pack WMMA operands (zero-pad K=3 -> 4, pair-interleave W1) ----
    pack_x4_k<<<(N * 4 + TB - 1) / TB, TB, 0, stream>>>(x, x4, N);
    pack_w1_k<<<4, TB, 0, stream>>>(W1, W1p);

    // ================= layer 1: H=4, C=64 =================
    gemm1_wmma<<<mtiles, TB, 0, stream>>>(x4, W1p, h1, N);
    att_dots_k<4, 64><<<(N * 4 + TB - 1) / TB, TB, 0, stream>>>(h1, aS1, aD1, a1s, a1d, N);
    edge_max_k<4><<<(Etot * 4 + TB - 1) / TB, TB, 0, stream>>>(ei, E, Etot, a1s, a1d, emax1);
    edge_den_k<4><<<(Etot * 4 + TB - 1) / TB, TB, 0, stream>>>(ei, E, Etot, a1s, a1d, emax1, den1);
    edge_agg_k<4, 64><<<(Etot + 7) / 8, TB, 0, stream>>>(ei, E, Etot, a1s, a1d, emax1, den1,
                                                         h1, agg1);
    elu_bias_k<<<2048, TB, 0, stream>>>(agg1, b1, N, 256);

    // ================= layer 2: H=1, C=128 =================
    gemm2_wmma<<<mtiles, TB, 0, stream>>>(agg1, W2, h2, N);
    att_dots_k<1, 128><<<(N + TB - 1) / TB, TB, 0, stream>>>(h2, aS2, aD2, a2s, a2d, N);
    edge_max_k<1><<<(Etot + TB - 1) / TB, TB, 0, stream>>>(ei, E, Etot, a2s, a2d, emax2);
    edge_den_k<1><<<(Etot + TB - 1) / TB, TB, 0, stream>>>(ei, E, Etot, a2s, a2d, emax2, den2);
    edge_agg_k<1, 128><<<(Etot + 7) / 8, TB, 0, stream>>>(ei, E, Etot, a2s, a2d, emax2, den2,
                                                          h2, out);
    bias_add_k<<<2048, TB, 0, stream>>>(out, b2, N, 128);
}